// AEGConv2d_55825984913796
// MI455X (gfx1250) — compile-verified
//
#include <hip/hip_runtime.h>
#include <math.h>

typedef float v2f __attribute__((ext_vector_type(2)));
typedef float v8f __attribute__((ext_vector_type(8)));

#define CIN  32
#define COUT 64
#define HDIM 64
#define WDIM 64
#define NIMG 4

// Workspace layout (floats):
//  [0, 49152)      : Waeg pairs  (((p*32+cin)*6 + pr)*64 + co)*2 + e   (k = 2*pr+e, zero for k>=9)
//  [49152, 73728)  : Wconv pairs ((cin*6 + pr)*64 + co)*2 + e
#define WS_CONV_OFF 49152

// ---------------------------------------------------------------------------
// Stage A: pack weights.
//  W_k(co,cin,p)  = y_k * prod_{j>k, (p+j)%2==0} y_j        (aeg closed form)
//  CW_k(co,cin)   = conv_w[co][cin][k]                      (plain im2col A)
// ---------------------------------------------------------------------------
__global__ __launch_bounds__(256) void aeg_prep_kernel(
    const float* __restrict__ weight, const float* __restrict__ conv_w,
    float* __restrict__ ws) {
  int idx = blockIdx.x * blockDim.x + threadIdx.x;
  if (idx < 2 * CIN * COUT) {
    int co  = idx & 63;
    int cin = (idx >> 6) & 31;
    int p   = idx >> 11;
    float y[9];
#pragma unroll
    for (int k = 0; k < 9; ++k) y[k] = weight[(co * CIN + cin) * 9 + k];
    float wv[12];
    wv[9] = wv[10] = wv[11] = 0.0f;
    float sy = 1.0f;
#pragma unroll
    for (int k = 8; k >= 0; --k) {
      wv[k] = y[k] * sy;
      if (((p + k) & 1) == 0) sy *= y[k];
    }
#pragma unroll
    for (int k = 0; k < 12; ++k)
      ws[(((p * CIN + cin) * 6 + (k >> 1)) * 64 + co) * 2 + (k & 1)] = wv[k];
  } else if (idx < 2 * CIN * COUT + CIN * COUT) {
    int t   = idx - 2 * CIN * COUT;
    int co  = t & 63;
    int cin = t >> 6;
#pragma unroll
    for (int k = 0; k < 12; ++k) {
      float v = (k < 9) ? conv_w[(co * CIN + cin) * 9 + k] : 0.0f;
      ws[WS_CONV_OFF + ((cin * 6 + (k >> 1)) * 64 + co) * 2 + (k & 1)] = v;
    }
  }
}

// ---------------------------------------------------------------------------
// Per-wave accumulation, parity P as compile-time constant. One wave owns a
// 64(co) x 16(px) single-parity slab: 8 fp32 16x16 accumulators. B fragments
// are built once per cin and reused by 24 WMMAs (4 co-blocks x (aeg+conv) x 3
// K-chunks); only A fragments differ per co-block (immediate-offset b64
// loads, cb stride 128B, pair stride 512B, cin stride 3KB).
// ---------------------------------------------------------------------------
template <int P>
__device__ __forceinline__ void aeg_accum(
    const float* __restrict__ xn, const v2f* __restrict__ wa,
    const v2f* __restrict__ wc, int t, int h, bool hb,
    int o0, int o1, int o2, int o3, int o4, int o5, int o6, int o7, int o8,
    float m0, float m1, float m2, float m3, float m5, float m6, float m7,
    float m8, v8f (&accA)[4], v8f (&accC)[4]) {
#pragma unroll 2
  for (int cin = 0; cin < CIN; ++cin) {
    const float* xc = xn + cin * (HDIM * WDIM);
    // Unconditional clamped loads + multiplicative border masks.
    float x0 = xc[o0] * m0, x1 = xc[o1] * m1, x2 = xc[o2] * m2;
    float x3 = xc[o3] * m3, x4 = xc[o4],      x5 = xc[o5] * m5;
    float x6 = xc[o6] * m6, x7 = xc[o7] * m7, x8 = xc[o8] * m8;

    // suffix products q_k = x_k * prod_{j>k,(P+j) odd} x_j  (P constexpr)
    float q0, q1, q2, q3, q4, q5, q6, q7, q8;
    if (P == 0) {                      // sx picks up x_k at odd k
      q8 = x8;
      q7 = x7;            float s1 = x7;
      q6 = x6 * s1;
      q5 = x5 * s1;       float s2 = x5 * s1;
      q4 = x4 * s2;
      q3 = x3 * s2;       float s3 = x3 * s2;
      q2 = x2 * s3;
      q1 = x1 * s3;       float s4 = x1 * s3;
      q0 = x0 * s4;
    } else {                           // sx picks up x_k at even k
      q8 = x8;            float s1 = x8;
      q7 = x7 * s1;
      q6 = x6 * s1;       float s2 = x6 * s1;
      q5 = x5 * s2;
      q4 = x4 * s2;       float s3 = x4 * s2;
      q3 = x3 * s3;
      q2 = x2 * s3;       float s4 = x2 * s3;
      q1 = x1 * s4;
      q0 = x0 * s4;
    }

    // B fragments: lane N=t; VGPR_i holds K rows {4cc+i, 4cc+i+2} by half.
    v2f bA0; bA0.x = hb ? q2 : q0;   bA0.y = hb ? q3 : q1;
    v2f bA1; bA1.x = hb ? q6 : q4;   bA1.y = hb ? q7 : q5;
    v2f bA2; bA2.x = hb ? 0.0f : q8; bA2.y = 0.0f;
    v2f bC0; bC0.x = hb ? x2 : x0;   bC0.y = hb ? x3 : x1;
    v2f bC1; bC1.x = hb ? x6 : x4;   bC1.y = hb ? x7 : x5;
    v2f bC2; bC2.x = hb ? 0.0f : x8; bC2.y = 0.0f;

    const v2f* wap = wa + (size_t)cin * 6 * 64;
    const v2f* wcp = wc + (size_t)cin * 6 * 64;
#pragma unroll
    for (int cb = 0; cb < 4; ++cb) {
      const int am = cb * 16 + t;      // A row m for this co-block
      v2f aF0 = wap[(0 + h) * 64 + am];
      v2f aF1 = wap[(2 + h) * 64 + am];
      v2f aF2 = wap[(4 + h) * 64 + am];
      v2f cF0 = wcp[(0 + h) * 64 + am];
      v2f cF1 = wcp[(2 + h) * 64 + am];
      v2f cF2 = wcp[(4 + h) * 64 + am];
      accA[cb] = __builtin_amdgcn_wmma_f32_16x16x4_f32(
          false, aF0, false, bA0, (short)0, accA[cb], false, false);
      accC[cb] = __builtin_amdgcn_wmma_f32_16x16x4_f32(
          false, cF0, false, bC0, (short)0, accC[cb], false, false);
      accA[cb] = __builtin_amdgcn_wmma_f32_16x16x4_f32(
          false, aF1, false, bA1, (short)0, accA[cb], false, false);
      accC[cb] = __builtin_amdgcn_wmma_f32_16x16x4_f32(
          false, cF1, false, bC1, (short)0, accC[cb], false, false);
      accA[cb] = __builtin_amdgcn_wmma_f32_16x16x4_f32(
          false, aF2, false, bA2, (short)0, accA[cb], false, false);
      accC[cb] = __builtin_amdgcn_wmma_f32_16x16x4_f32(
          false, cF2, false, bC2, (short)0, accC[cb], false, false);
    }
  }
}

// ---------------------------------------------------------------------------
// Main kernel: block = one (n, row i); 4 wave32's: (strip jb, parity pw).
// Each wave computes all 64 co for its 16 same-parity pixels.
// ---------------------------------------------------------------------------
__global__ __launch_bounds__(128) void aeg_main_kernel(
    const float* __restrict__ x, const float* __restrict__ conv_b,
    const float* __restrict__ ws, float* __restrict__ out) {
  const int tid = threadIdx.x;
  const int b   = blockIdx.x;          // 256 blocks = 4n * 64 rows
  const int n   = b >> 6;
  const int i   = b & 63;              // block-uniform row

  const int w  = tid >> 5;             // wave 0..3 (wave32)
  const int jb = w >> 1;               // 32-col strip
  const int pw = w & 1;                // parity selector within strip
  const int t  = tid & 15;             // tile column  (N = lane&15)
  const bool hb = (tid & 16) != 0;     // lane half    (selects K rows)
  const int h  = hb ? 1 : 0;

  const int j = jb * 32 + 2 * t + pw;  // this lane's pixel column
  // pixel parity (i+j)&1 == (i+pw)&1 : wave-uniform -> force to SGPR
  const int p = __builtin_amdgcn_readfirstlane((i + pw) & 1);

  // Loop-invariant clamped tap offsets (elements) + zero masks.
  const int ir0 = (i > 0) ? i - 1 : 0;
  const int ir2 = (i < HDIM - 1) ? i + 1 : HDIM - 1;
  const int jc0 = (j > 0) ? j - 1 : 0;
  const int jc2 = (j < WDIM - 1) ? j + 1 : WDIM - 1;
  const float mr0 = (i > 0) ? 1.0f : 0.0f;          // scalar
  const float mr2 = (i < HDIM - 1) ? 1.0f : 0.0f;   // scalar
  const float mc0 = (j > 0) ? 1.0f : 0.0f;          // per-lane
  const float mc2 = (j < WDIM - 1) ? 1.0f : 0.0f;   // per-lane
  const float m0 = mr0 * mc0, m1 = mr0, m2 = mr0 * mc2;
  const float m3 = mc0,                 m5 = mc2;
  const float m6 = mr2 * mc0, m7 = mr2, m8 = mr2 * mc2;
  const int o0 = ir0 * WDIM + jc0, o1 = ir0 * WDIM + j, o2 = ir0 * WDIM + jc2;
  const int o3 = i   * WDIM + jc0, o4 = i   * WDIM + j, o5 = i   * WDIM + jc2;
  const int o6 = ir2 * WDIM + jc0, o7 = ir2 * WDIM + j, o8 = ir2 * WDIM + jc2;

  v8f accA[4] = {{}, {}, {}, {}};      // aeg accumulators, cb = 0..3
  v8f accC[4] = {{}, {}, {}, {}};      // conv accumulators

  const float* xn = x + (size_t)n * CIN * HDIM * WDIM;
  const v2f* wa = (const v2f*)ws + (size_t)(p * CIN) * 6 * 64;
  const v2f* wc = (const v2f*)(ws + WS_CONV_OFF);

  if (p == 0)
    aeg_accum<0>(xn, wa, wc, t, h, hb, o0, o1, o2, o3, o4, o5, o6, o7, o8,
                 m0, m1, m2, m3, m5, m6, m7, m8, accA, accC);
  else
    aeg_accum<1>(xn, wa, wc, t, h, hb, o0, o1, o2, o3, o4, o5, o6, o7, o8,
                 m0, m1, m2, m3, m5, m6, m7, m8, accA, accC);

  // Epilogue: D layout -> lane covers (co = cb*16 + v + 8h, pixel j)
#pragma unroll
  for (int cb = 0; cb < 4; ++cb)
#pragma unroll
    for (int v = 0; v < 8; ++v) {
      int co = cb * 16 + v + 8 * h;
      float aeg  = accA[cb][v];
      float conv = accC[cb][v] + conv_b[co];
      float sig  = 1.0f / (1.0f + expf(-aeg));
      out[(((size_t)n * COUT + co) * HDIM + i) * WDIM + j] = sig * conv;
    }
}

extern "C" void kernel_launch(void* const* d_in, const int* in_sizes, int n_in,
                              void* d_out, int out_size, void* d_ws, size_t ws_size,
                              hipStream_t stream) {
  const float* x      = (const float*)d_in[0];
  const float* weight = (const float*)d_in[1];
  const float* conv_w = (const float*)d_in[2];
  const float* conv_b = (const float*)d_in[3];
  float* ws  = (float*)d_ws;
  float* out = (float*)d_out;

  // Stage A: 2*32*64 (aeg, per parity) + 32*64 (conv) = 6144 packing threads.
  aeg_prep_kernel<<<24, 256, 0, stream>>>(weight, conv_w, ws);
  // Main: 256 blocks (n, row) x 128 threads (4 wave32 slabs: strip x parity).
  aeg_main_kernel<<<256, 128, 0, stream>>>(x, conv_b, ws, out);
}